// TextGuidedSampler_49572512530550
// MI455X (gfx1250) — compile-verified
//
#include <hip/hip_runtime.h>
#include <hip/hip_bf16.h>

// ---------------------------------------------------------------------------
// TextGuidedSampler for MI455X (gfx1250, wave32, WMMA f32 16x16x4)
// B=16, NV=4096, NT=64, D=1024, VISION_TOPK=32, TEXT_TOPK=5
// ---------------------------------------------------------------------------

#define B_   16
#define NV_  4096
#define NT_  64
#define D_   1024
#define VK_  32
#define TK_  5
#define NEG_INF (-3.402823466e38f)

typedef __attribute__((ext_vector_type(2))) float v2f;
typedef __attribute__((ext_vector_type(8))) float v8f;

// -------------------------------------------------------------------- rnorm
// out[row] = 1 / max(||src[row,:]||_2, 1e-12), D = 1024, block = 256
__global__ __launch_bounds__(256) void tg_rnorm_kernel(
    const float* __restrict__ src, float* __restrict__ out) {
  const int row = blockIdx.x;
  const int t = threadIdx.x;
  const float4 v = ((const float4*)(src + (size_t)row * D_))[t];
  float s = v.x * v.x + v.y * v.y + v.z * v.z + v.w * v.w;
  __shared__ float red[256];
  red[t] = s;
  __syncthreads();
  for (int off = 128; off > 0; off >>= 1) {
    if (t < off) red[t] += red[t + off];
    __syncthreads();
  }
  if (t == 0) out[row] = 1.0f / fmaxf(sqrtf(red[0]), 1e-12f);
}

// ---------------------------------------------------------------- sim WMMA
// sim[b, r0+m, n] = <normV[b,r0+m,:], normT[b,n,:]> , masked -> -1.0
// grid = (NV/64, B), block = 256 (8 waves). Each block: 64x64 tile of sim.
// Each wave owns two 16x16 output tiles; K staged through LDS in 32-chunks.
__global__ __launch_bounds__(256) void tg_sim_kernel(
    const float* __restrict__ V, const float* __restrict__ T,
    const float* __restrict__ rnV, const float* __restrict__ rnT,
    const unsigned char* __restrict__ mask, float* __restrict__ sim) {
  const int b = blockIdx.y;
  const int r0 = blockIdx.x * 64;
  const int tid = threadIdx.x;
  const int lane = tid & 31, wave = tid >> 5;
  const int half = lane >> 4, idx = lane & 15;

  // 64 rows x 32 K, stride 40 floats (160B -> float4-aligned, staggered banks)
  __shared__ float As[64 * 40];
  __shared__ float Bs[64 * 40];

  const int tile0 = wave;      // tiles 0..7
  const int tile1 = wave + 8;  // tiles 8..15
  const int tm0 = tile0 >> 2, tn0 = tile0 & 3;
  const int tm1 = tile1 >> 2, tn1 = tile1 & 3;

  v8f acc0 = {};
  v8f acc1 = {};

  for (int k0 = 0; k0 < D_; k0 += 32) {
    // stage normalized A (vision) and B (text) chunks, coalesced float4 loads
    for (int f = tid; f < 512; f += 256) {
      const int row = f >> 3, c4 = (f & 7) << 2;
      const float rn = rnV[b * NV_ + r0 + row];
      const float4 v =
          *(const float4*)(V + ((size_t)(b * NV_ + r0 + row)) * D_ + k0 + c4);
      float* dst = &As[row * 40 + c4];
      dst[0] = v.x * rn; dst[1] = v.y * rn; dst[2] = v.z * rn; dst[3] = v.w * rn;
    }
    for (int f = tid; f < 512; f += 256) {
      const int row = f >> 3, c4 = (f & 7) << 2;
      const float rn = rnT[b * NT_ + row];
      const float4 v =
          *(const float4*)(T + ((size_t)(b * NT_ + row)) * D_ + k0 + c4);
      float* dst = &Bs[row * 40 + c4];
      dst[0] = v.x * rn; dst[1] = v.y * rn; dst[2] = v.z * rn; dst[3] = v.w * rn;
    }
    __syncthreads();

#pragma unroll
    for (int kk = 0; kk < 32; kk += 4) {
      const int ka = kk + 2 * half;  // lanes 0-15: K=kk,kk+1; lanes 16-31: K=kk+2,kk+3
      v2f a, bb;
      a.x = As[(tm0 * 16 + idx) * 40 + ka];
      a.y = As[(tm0 * 16 + idx) * 40 + ka + 1];
      bb.x = Bs[(tn0 * 16 + idx) * 40 + ka];
      bb.y = Bs[(tn0 * 16 + idx) * 40 + ka + 1];
      acc0 = __builtin_amdgcn_wmma_f32_16x16x4_f32(false, a, false, bb,
                                                   (short)0, acc0, false, false);
      a.x = As[(tm1 * 16 + idx) * 40 + ka];
      a.y = As[(tm1 * 16 + idx) * 40 + ka + 1];
      bb.x = Bs[(tn1 * 16 + idx) * 40 + ka];
      bb.y = Bs[(tn1 * 16 + idx) * 40 + ka + 1];
      acc1 = __builtin_amdgcn_wmma_f32_16x16x4_f32(false, a, false, bb,
                                                   (short)0, acc1, false, false);
    }
    __syncthreads();
  }

  // store (masked text columns -> -1.0). C/D layout: lane<16: M=i; lane>=16: M=i+8
  {
    int n = tn0 * 16 + idx;
    const bool ok0 = mask[b * NT_ + n] != 0;
#pragma unroll
    for (int i = 0; i < 8; i++) {
      const int m = r0 + tm0 * 16 + i + 8 * half;
      sim[((size_t)(b * NV_ + m)) * NT_ + n] = ok0 ? acc0[i] : -1.0f;
    }
    n = tn1 * 16 + idx;
    const bool ok1 = mask[b * NT_ + n] != 0;
#pragma unroll
    for (int i = 0; i < 8; i++) {
      const int m = r0 + tm1 * 16 + i + 8 * half;
      sim[((size_t)(b * NV_ + m)) * NT_ + n] = ok1 ? acc1[i] : -1.0f;
    }
  }
}

// ------------------------------------------------------------ mean over text
__global__ __launch_bounds__(256) void tg_vprobs_kernel(
    const float* __restrict__ sim, float* __restrict__ vp) {
  const int g = blockIdx.x * 256 + threadIdx.x;  // 0..B*NV-1
  const float* p = sim + (size_t)g * NT_;
  float s = 0.0f;
#pragma unroll 8
  for (int t = 0; t < NT_; t++) s += p[t];
  vp[g] = s * (1.0f / (float)NT_);
}

// ---------------------------------------------------------- mean over vision
__global__ __launch_bounds__(256) void tg_tprobs_kernel(
    const float* __restrict__ sim, float* __restrict__ tp) {
  const int b = blockIdx.x >> 6;
  const int t = blockIdx.x & 63;
  float s = 0.0f;
  for (int n = threadIdx.x; n < NV_; n += 256)
    s += sim[((size_t)(b * NV_ + n)) * NT_ + t];
  __shared__ float red[256];
  red[threadIdx.x] = s;
  __syncthreads();
  for (int off = 128; off > 0; off >>= 1) {
    if (threadIdx.x < off) red[threadIdx.x] += red[threadIdx.x + off];
    __syncthreads();
  }
  if (threadIdx.x == 0) tp[blockIdx.x] = red[0] * (1.0f / (float)NV_);
}

// ------------------------------------------------- vision top-32 (stable)
__global__ __launch_bounds__(256) void tg_topkv_kernel(
    const float* __restrict__ vp, int* __restrict__ vidx) {
  const int b = blockIdx.x;
  const int tid = threadIdx.x;
  __shared__ float buf[NV_];
  __shared__ float bv[256];
  __shared__ int bi[256];
  __shared__ int picks[VK_];
  for (int i = tid; i < NV_; i += 256) buf[i] = vp[b * NV_ + i];
  __syncthreads();
  for (int it = 0; it < VK_; it++) {
    float best = NEG_INF;
    int besti = NV_;
    for (int i = tid; i < NV_; i += 256) {
      const float v = buf[i];
      if (v > best || (v == best && i < besti)) { best = v; besti = i; }
    }
    bv[tid] = best;
    bi[tid] = besti;
    __syncthreads();
    if (tid == 0) {
      float bb = bv[0];
      int bbi = bi[0];
      for (int i = 1; i < 256; i++)
        if (bv[i] > bb || (bv[i] == bb && bi[i] < bbi)) { bb = bv[i]; bbi = bi[i]; }
      picks[it] = bbi;
      buf[bbi] = NEG_INF;  // remove
    }
    __syncthreads();
  }
  if (tid == 0) {  // ascending index sort + write
    for (int i = 1; i < VK_; i++) {
      const int key = picks[i];
      int k = i - 1;
      while (k >= 0 && picks[k] > key) { picks[k + 1] = picks[k]; k--; }
      picks[k + 1] = key;
    }
    for (int i = 0; i < VK_; i++) vidx[b * VK_ + i] = picks[i];
  }
}

// --------------------------------------------------- text top-5 (stable)
__global__ void tg_topkt_kernel(const float* __restrict__ tp,
                                int* __restrict__ tidx) {
  if (threadIdx.x != 0) return;
  const int b = blockIdx.x;
  float v[NT_];
  for (int i = 0; i < NT_; i++) v[i] = tp[b * NT_ + i];
  int p[TK_];
  for (int it = 0; it < TK_; it++) {
    float best = NEG_INF;
    int bi2 = 0;
    for (int i = 0; i < NT_; i++)
      if (v[i] > best) { best = v[i]; bi2 = i; }  // strict > keeps lowest index
    p[it] = bi2;
    v[bi2] = NEG_INF;
  }
  for (int i = 0; i < TK_; i++)  // ascending
    for (int j = i + 1; j < TK_; j++)
      if (p[j] < p[i]) { const int tmp = p[i]; p[i] = p[j]; p[j] = tmp; }
  for (int i = 0; i < TK_; i++) tidx[b * TK_ + i] = p[i];
}

// --------------------------- softmax over vision axis for selected text rows
// attn[b, j, n] = softmax_n(sim[b, n, t_idx[b,j]])   grid = (TK, B)
__global__ __launch_bounds__(256) void tg_softmax_kernel(
    const float* __restrict__ sim, const int* __restrict__ tidx,
    float* __restrict__ attn) {
  const int b = blockIdx.y, j = blockIdx.x;
  const int t = tidx[b * TK_ + j];
  const int tid = threadIdx.x;
  __shared__ float red[256];
  __shared__ float s_mx, s_inv;

  float mx = NEG_INF;
  for (int n = tid; n < NV_; n += 256)
    mx = fmaxf(mx, sim[((size_t)(b * NV_ + n)) * NT_ + t]);
  red[tid] = mx;
  __syncthreads();
  for (int off = 128; off > 0; off >>= 1) {
    if (tid < off) red[tid] = fmaxf(red[tid], red[tid + off]);
    __syncthreads();
  }
  if (tid == 0) s_mx = red[0];
  __syncthreads();
  mx = s_mx;

  float se = 0.0f;
  for (int n = tid; n < NV_; n += 256)
    se += __expf(sim[((size_t)(b * NV_ + n)) * NT_ + t] - mx);
  __syncthreads();
  red[tid] = se;
  __syncthreads();
  for (int off = 128; off > 0; off >>= 1) {
    if (tid < off) red[tid] += red[tid + off];
    __syncthreads();
  }
  if (tid == 0) s_inv = 1.0f / red[0];
  __syncthreads();
  const float inv = s_inv;

  for (int n = tid; n < NV_; n += 256)
    attn[((size_t)(b * TK_ + j)) * NV_ + n] =
        __expf(sim[((size_t)(b * NV_ + n)) * NT_ + t] - mx) * inv;
}

// ------------------------- weighted vision via WMMA: out = attn_sel @ vision
// A = attn rows (M=16, rows 0..4 valid), B = vision [K x D] tiles, K = 4096.
// grid = (8, B), block = 256 (8 waves) -> each wave owns one 16-wide D tile;
// 8 x-blocks * 8 waves => grid.x = 8, 64 D-tiles total.
__global__ __launch_bounds__(256) void tg_weighted_kernel(
    const float* __restrict__ V, const float* __restrict__ attn,
    float* __restrict__ out) {
  const int b = blockIdx.y;
  const int tid = threadIdx.x;
  const int lane = tid & 31, wave = tid >> 5;
  const int half = lane >> 4, idx = lane & 15;
  const int d0 = (blockIdx.x * 8 + wave) * 16;

  __shared__ float Aw[TK_ * 256];  // 5 rows x 256-K chunk, shared by all waves

  v8f acc = {};
  for (int k0 = 0; k0 < NV_; k0 += 256) {
    for (int f = tid; f < TK_ * 256; f += 256) {
      const int row = f >> 8, c = f & 255;
      Aw[row * 256 + c] = attn[((size_t)(b * TK_ + row)) * NV_ + k0 + c];
    }
    __syncthreads();
#pragma unroll 4
    for (int kk = 0; kk < 256; kk += 4) {
      const int ka = kk + 2 * half;
      v2f a, bb;
      a.x = (idx < TK_) ? Aw[idx * 256 + ka] : 0.0f;
      a.y = (idx < TK_) ? Aw[idx * 256 + ka + 1] : 0.0f;
      const size_t base = ((size_t)(b * NV_) + k0 + ka) * D_ + d0 + idx;
      bb.x = V[base];
      bb.y = V[base + D_];
      acc = __builtin_amdgcn_wmma_f32_16x16x4_f32(false, a, false, bb,
                                                  (short)0, acc, false, false);
    }
    __syncthreads();
  }
  // rows 0..4 live in VGPRs 0..4 of lanes 0..15 (M = i for lane<16)
  if (lane < 16) {
#pragma unroll
    for (int i = 0; i < TK_; i++)
      out[((size_t)(b * (TK_ + VK_)) + i) * D_ + d0 + lane] = acc[i];
  }
}

// ------------------------------------------------ gather top-32 vision rows
__global__ __launch_bounds__(256) void tg_gather_kernel(
    const float* __restrict__ V, const int* __restrict__ vidx,
    float* __restrict__ out) {
  const int b = blockIdx.y, j = blockIdx.x;
  const int r = vidx[b * VK_ + j];
  const float4* src = (const float4*)(V + ((size_t)(b * NV_ + r)) * D_);
  float4* dst =
      (float4*)(out + ((size_t)(b * (TK_ + VK_)) + TK_ + j) * D_);
  dst[threadIdx.x] = src[threadIdx.x];  // 256 * float4 = 1024 floats
}

// ---------------------------------------------------------------------------
extern "C" void kernel_launch(void* const* d_in, const int* in_sizes, int n_in,
                              void* d_out, int out_size, void* d_ws,
                              size_t ws_size, hipStream_t stream) {
  const float* vision = (const float*)d_in[0];          // [B,NV,D] f32
  const float* text = (const float*)d_in[1];            // [B,NT,D] f32
  const unsigned char* mask = (const unsigned char*)d_in[2];  // [B,NT] bool
  float* out = (float*)d_out;                           // [B, 37, D] f32

  float* ws = (float*)d_ws;
  float* rnV = ws;                                   // B*NV
  float* rnT = rnV + (size_t)B_ * NV_;               // B*NT
  float* sim = rnT + (size_t)B_ * NT_;               // B*NV*NT
  float* vp = sim + (size_t)B_ * NV_ * NT_;          // B*NV
  float* tp = vp + (size_t)B_ * NV_;                 // B*NT
  float* attn = tp + (size_t)B_ * NT_;               // B*TK*NV
  int* vidx = (int*)(attn + (size_t)B_ * TK_ * NV_); // B*VK
  int* tidx = vidx + (size_t)B_ * VK_;               // B*TK

  tg_rnorm_kernel<<<B_ * NV_, 256, 0, stream>>>(vision, rnV);
  tg_rnorm_kernel<<<B_ * NT_, 256, 0, stream>>>(text, rnT);

  tg_sim_kernel<<<dim3(NV_ / 64, B_), 256, 0, stream>>>(vision, text, rnV, rnT,
                                                        mask, sim);

  tg_vprobs_kernel<<<(B_ * NV_) / 256, 256, 0, stream>>>(sim, vp);
  tg_tprobs_kernel<<<B_ * NT_, 256, 0, stream>>>(sim, tp);

  tg_topkv_kernel<<<B_, 256, 0, stream>>>(vp, vidx);
  tg_topkt_kernel<<<B_, 64, 0, stream>>>(tp, tidx);

  tg_softmax_kernel<<<dim3(TK_, B_), 256, 0, stream>>>(sim, tidx, attn);

  tg_weighted_kernel<<<dim3(8, B_), 256, 0, stream>>>(vision, attn, out);
  tg_gather_kernel<<<dim3(VK_, B_), 256, 0, stream>>>(vision, vidx, out);
}